// OctreeDecoder_15479062134892
// MI455X (gfx1250) — compile-verified
//
#include <hip/hip_runtime.h>
#include <math.h>

typedef __attribute__((ext_vector_type(2))) float v2f;
typedef __attribute__((ext_vector_type(8))) float v8f;

// ---------------------------------------------------------------------------
// Level 0 init: x0 = x * occ0, occ_f = float(occ0)
// ---------------------------------------------------------------------------
__global__ void k_init(const float* __restrict__ xin, const int* __restrict__ occ0,
                       float* __restrict__ x0, float* __restrict__ occf,
                       int npix, int C) {
    int idx = blockIdx.x * blockDim.x + threadIdx.x;
    if (idx >= npix * C) return;
    int pix = idx / C;
    float o = (float)occ0[pix];
    x0[idx] = xin[idx] * o;
    if ((idx % C) == 0) occf[pix] = o;
}

// ---------------------------------------------------------------------------
// Occupancy expansion: up[2i]=occ[i]; dilate 3x3 SAME (init 0).
// even o -> occ[o/2]; odd o -> max(occ[(o-1)/2], occ[(o+1)/2] if in range)
// ---------------------------------------------------------------------------
__global__ void k_expand(const float* __restrict__ occ_in, float* __restrict__ occ_out,
                         int B, int Hin, int Win) {
    int Hout = 2 * Hin, Wout = 2 * Win;
    int idx = blockIdx.x * blockDim.x + threadIdx.x;
    int total = B * Hout * Wout;
    if (idx >= total) return;
    int ow = idx % Wout; int t = idx / Wout;
    int oh = t % Hout;   int b = t / Hout;
    int ih0, ih1, iw0, iw1;
    if (oh & 1) { ih0 = oh >> 1; ih1 = (ih0 + 1 < Hin) ? ih0 + 1 : ih0; }
    else        { ih0 = ih1 = oh >> 1; }
    if (ow & 1) { iw0 = ow >> 1; iw1 = (iw0 + 1 < Win) ? iw0 + 1 : iw0; }
    else        { iw0 = iw1 = ow >> 1; }
    const float* p = occ_in + (size_t)b * Hin * Win;
    float v = fmaxf(fmaxf(p[ih0 * Win + iw0], p[ih0 * Win + iw1]),
                    fmaxf(p[ih1 * Win + iw0], p[ih1 * Win + iw1]));
    occ_out[idx] = v;
}

// ---------------------------------------------------------------------------
// Transposed conv (k=3, s=2, SAME, HWIO, no kernel flip) as implicit GEMM on
// V_WMMA_F32_16X16X4_F32.  One wave32 per tile of (16 same-parity output
// pixels) x (16 output channels).  CIN/COUT are compile-time so the K loop
// unrolls into clauses of b64/b32 loads with immediate offsets feeding
// back-to-back WMMAs.  Left-edge taps are handled by clamped-address loads
// multiplied by a {0,1} lane mask (no EXEC divergence around WMMA).
// Output: relu(conv + bias) * occ_out   (fused masking)
// ---------------------------------------------------------------------------
template <int CIN, int COUT>
__global__ __launch_bounds__(256)
void k_deconv(const float* __restrict__ xin, const float* __restrict__ wgt,
              const float* __restrict__ bias, const float* __restrict__ occo,
              float* __restrict__ xout, int B, int Hin) {
    const int Win = Hin;
    const int Hout = 2 * Hin, Wout = 2 * Win;
    const int lane = threadIdx.x & 31;
    const int m    = lane & 15;      // A-row / B-col index
    const int half = lane >> 4;      // K-pair select (A/B), M offset (C/D)
    const int kofs = 2 * half;       // ISA 16x4 f32 A layout: K = 2h, 2h+1

    const int ntn  = COUT >> 4;          // N tiles
    const int ngrp = Wout >> 4;          // 16-pixel same-parity groups / row
    long wid = (long)((blockIdx.x * blockDim.x + threadIdx.x) >> 5);
    long total = (long)B * Hout * ngrp * ntn;
    if (wid >= total) return;

    int nt  = (int)(wid % ntn);  long t = wid / ntn;
    int owg = (int)(t % ngrp);   t /= ngrp;
    int oh  = (int)(t % Hout);
    int b   = (int)(t / Hout);
    int p = owg & 1, g = owg >> 1;
    int iwbase = 16 * g + m;             // per-lane input column base

    v8f acc = {};

#pragma unroll
    for (int th = 0; th < 2; ++th) {
        int dk_h, ih; bool hok;
        if (oh & 1) { hok = (th == 0); dk_h = 1; ih = oh >> 1; }
        else        { dk_h = th * 2; ih = (oh >> 1) + th - 1; hok = (ih >= 0); }
        if (!hok) continue;                              // wave-uniform
#pragma unroll
        for (int tw = 0; tw < 2; ++tw) {
            int dk_w, iwof; bool wok;
            if (p) { wok = (tw == 0); dk_w = 1; iwof = 0; }
            else   { wok = true; dk_w = tw * 2; iwof = tw - 1; }
            if (!wok) continue;                          // wave-uniform

            int iw = iwbase + iwof;
            float msk = (iw >= 0) ? 1.0f : 0.0f;         // per-lane edge mask
            const float* xp = xin +
                ((size_t)(b * Hin + ih) * Win + (iw >= 0 ? iw : 0)) * CIN + kofs;
            const float* wp = wgt +
                (size_t)(dk_h * 3 + dk_w) * CIN * COUT +
                (size_t)kofs * COUT + nt * 16 + m;

#pragma unroll 8
            for (int kc = 0; kc < CIN; kc += 4) {
                v2f a = *(const v2f*)(xp + kc);          // b64, 8B aligned
                a.x *= msk;  a.y *= msk;
                v2f bb;
                bb.x = wp[(size_t)kc * COUT];            // row kofs   (+imm)
                bb.y = wp[(size_t)kc * COUT + COUT];     // row kofs+1 (+imm)
                acc = __builtin_amdgcn_wmma_f32_16x16x4_f32(
                          false, a, false, bb, (short)0, acc, false, false);
            }
        }
    }

    int co = nt * 16 + m;
    float bv = bias[co];
#pragma unroll
    for (int r = 0; r < 8; ++r) {
        int mp = r + 8 * half;                           // C/D: M = r + 8*half
        int ow = p + 2 * (16 * g + mp);
        size_t pix = (size_t)(b * Hout + oh) * Wout + ow;
        float o = fmaxf(acc[r] + bv, 0.0f) * occo[pix];
        xout[pix * COUT + co] = o;
    }
}

// ---------------------------------------------------------------------------
// Per-pixel head: out = sigmoid(x.out_w + b)*occ ; label MLP ; prune occ & x.
// One wave32 per pixel; HIDDEN==32 maps to lanes.
// ---------------------------------------------------------------------------
__global__ void k_head(float* __restrict__ x, float* __restrict__ occ,
                       const float* __restrict__ ow_, const float* __restrict__ ob_,
                       const float* __restrict__ w1, const float* __restrict__ b1,
                       const float* __restrict__ w2, const float* __restrict__ b2,
                       float* __restrict__ out, int npix, int C) {
    int wid  = (blockIdx.x * blockDim.x + threadIdx.x) >> 5;
    int lane = threadIdx.x & 31;
    if (wid >= npix) return;
    float* xp  = x + (size_t)wid * C;
    float occv = occ[wid];

    // sigmoid head (scalar dot, lane-strided + xor reduce)
    float s = 0.0f;
    for (int c = lane; c < C; c += 32) s += xp[c] * ow_[c];
#pragma unroll
    for (int msk = 16; msk >= 1; msk >>= 1) s += __shfl_xor(s, msk, 32);
    float outv = occv / (1.0f + expf(-(s + ob_[0])));

    // hidden: lane j owns h[j]
    float h = b1[lane];
    for (int c = 0; c < C; ++c) h += xp[c] * w1[c * 32 + lane];
    h = fmaxf(h, 0.0f);

    float l0 = h * w2[lane * 2 + 0];
    float l1 = h * w2[lane * 2 + 1];
#pragma unroll
    for (int msk = 16; msk >= 1; msk >>= 1) {
        l0 += __shfl_xor(l0, msk, 32);
        l1 += __shfl_xor(l1, msk, 32);
    }
    l0 += b2[0]; l1 += b2[1];
    float label  = (l1 > l0) ? 1.0f : 0.0f;     // argmax, tie -> 0
    float newocc = occv * label;

    if (lane == 0) { out[wid] = outv; occ[wid] = newocc; }
    if (newocc == 0.0f) {
        for (int c = lane; c < C; c += 32) xp[c] = 0.0f;
    }
}

// ---------------------------------------------------------------------------
extern "C" void kernel_launch(void* const* d_in, const int* in_sizes, int n_in,
                              void* d_out, int out_size, void* d_ws, size_t ws_size,
                              hipStream_t stream) {
    (void)in_sizes; (void)n_in; (void)out_size; (void)ws_size;
    static const int CH[6] = {256, 128, 64, 48, 32, 16};
    const int B = 4;

    const float* x    = (const float*)d_in[0];
    const int*   occ0 = (const int*)d_in[1];
    const float *dw[5], *db[5], *ow[6], *ob[6], *w1[6], *b1[6], *w2[6], *b2[6];
    for (int i = 0; i < 5; ++i) { dw[i] = (const float*)d_in[2 + i];
                                  db[i] = (const float*)d_in[7 + i]; }
    for (int i = 0; i < 6; ++i) {
        ow[i] = (const float*)d_in[12 + i];
        ob[i] = (const float*)d_in[18 + i];
        w1[i] = (const float*)d_in[24 + i];
        b1[i] = (const float*)d_in[30 + i];
        w2[i] = (const float*)d_in[36 + i];
        b2[i] = (const float*)d_in[42 + i];
    }
    float* out = (float*)d_out;

    // workspace layout: per-level feature maps, then per-level float occ
    float* ws = (float*)d_ws;
    size_t off = 0;
    float* xs[6]; float* os[6];
    for (int d = 0; d < 6; ++d) {
        int Hd = 16 << d;
        xs[d] = ws + off; off += (size_t)B * Hd * Hd * CH[d];
    }
    for (int d = 0; d < 6; ++d) {
        int Hd = 16 << d;
        os[d] = ws + off; off += (size_t)B * Hd * Hd;
    }

    int npix0 = B * 16 * 16;
    k_init<<<(npix0 * CH[0] + 255) / 256, 256, 0, stream>>>(x, occ0, xs[0], os[0],
                                                            npix0, CH[0]);
    size_t oofs = 0;
    for (int d = 0; d < 6; ++d) {
        int Hd = 16 << d;
        int npix = B * Hd * Hd;
        int C = CH[d];
        if (d >= 1) {
            int Hin = Hd >> 1;
            k_expand<<<(npix + 255) / 256, 256, 0, stream>>>(os[d - 1], os[d],
                                                             B, Hin, Hin);
            long tiles = (long)B * Hd * (Hd >> 4) * (C >> 4);
            int blocks = (int)((tiles + 7) / 8);       // 8 waves / block
            switch (d) {
            case 1: k_deconv<256, 128><<<blocks, 256, 0, stream>>>(
                        xs[0], dw[0], db[0], os[1], xs[1], B, Hin); break;
            case 2: k_deconv<128, 64><<<blocks, 256, 0, stream>>>(
                        xs[1], dw[1], db[1], os[2], xs[2], B, Hin); break;
            case 3: k_deconv<64, 48><<<blocks, 256, 0, stream>>>(
                        xs[2], dw[2], db[2], os[3], xs[3], B, Hin); break;
            case 4: k_deconv<48, 32><<<blocks, 256, 0, stream>>>(
                        xs[3], dw[3], db[3], os[4], xs[4], B, Hin); break;
            case 5: k_deconv<32, 16><<<blocks, 256, 0, stream>>>(
                        xs[4], dw[4], db[4], os[5], xs[5], B, Hin); break;
            }
        }
        int hblocks = (npix + 7) / 8;                  // 8 waves / block, wave/pixel
        k_head<<<hblocks, 256, 0, stream>>>(xs[d], os[d], ow[d], ob[d],
                                            w1[d], b1[d], w2[d], b2[d],
                                            out + oofs, npix, C);
        oofs += (size_t)npix;
    }
}